// rosa_4bit_layer_84679575208362
// MI455X (gfx1250) — compile-verified
//
#include <hip/hip_runtime.h>

// out[b,t,c] = (x[b,t,c] > 0) ? emb1[c] : emb0[c]
// B=8, T=4096, C=1024  ->  rows = B*T = 32768, each row is C floats.
// Pure HBM-streaming kernel: ~256 MiB traffic, ~11.5 us floor at 23.3 TB/s.

typedef float f4  __attribute__((ext_vector_type(4)));
typedef int   v4i __attribute__((ext_vector_type(4)));

typedef __attribute__((address_space(1))) v4i* gvec_ptr;   // global
typedef __attribute__((address_space(3))) v4i* lvec_ptr;   // LDS

#define ROSA_C        1024
#define ROSA_ROWS     32768      // B*T
#define ROSA_RPB      16         // rows per block
#define ROSA_BLOCKS   (ROSA_ROWS / ROSA_RPB)   // 2048
#define ROSA_THREADS  (ROSA_C / 4)             // 256: thread t owns channels [4t,4t+4)

__global__ __launch_bounds__(ROSA_THREADS) void rosa_select_kernel(
    const float* __restrict__ x,
    const float* __restrict__ emb0,
    const float* __restrict__ emb1,
    float* __restrict__ out)
{
    __shared__ float s_emb[2 * ROSA_C];   // [0,1024): emb0, [1024,2048): emb1
    const int tid = threadIdx.x;

    // ---- Stage the 8 KB of embeddings into LDS via the CDNA5 async path ----
#if defined(__gfx1250__) && __has_builtin(__builtin_amdgcn_global_load_async_to_lds_b128)
    __builtin_amdgcn_global_load_async_to_lds_b128(
        (gvec_ptr)(emb0 + 4 * tid),
        (lvec_ptr)(s_emb + 4 * tid),
        /*offset=*/0, /*cpol=*/0);
    __builtin_amdgcn_global_load_async_to_lds_b128(
        (gvec_ptr)(emb1 + 4 * tid),
        (lvec_ptr)(s_emb + ROSA_C + 4 * tid),
        /*offset=*/0, /*cpol=*/0);
  #if __has_builtin(__builtin_amdgcn_s_wait_asynccnt)
    __builtin_amdgcn_s_wait_asynccnt(0);
  #else
    asm volatile("s_wait_asynccnt 0" ::: "memory");
  #endif
    __syncthreads();
#else
    // Fallback: plain cooperative copy through VGPRs.
    *(f4*)(s_emb + 4 * tid)          = *(const f4*)(emb0 + 4 * tid);
    *(f4*)(s_emb + ROSA_C + 4 * tid) = *(const f4*)(emb1 + 4 * tid);
    __syncthreads();
#endif

    // Hoist this thread's fixed 4-channel embedding pair into registers
    // (ds_load_b128 x2); reused for all ROSA_RPB rows.
    const f4 e0 = *(const f4*)(s_emb + 4 * tid);
    const f4 e1 = *(const f4*)(s_emb + ROSA_C + 4 * tid);

    // Contiguous 64 KB chunk of x/out per block: good DRAM page locality.
    long base = (long)blockIdx.x * (ROSA_RPB * ROSA_C) + 4 * tid;

#pragma unroll 4
    for (int r = 0; r < ROSA_RPB; ++r) {
        const long off = base + (long)r * ROSA_C;
        // Streaming data: non-temporal load/store (th:TH_*_NT) — x+out
        // (256 MiB) would otherwise thrash the 192 MB L2.
        f4 xv = __builtin_nontemporal_load((const f4*)(x + off));
        f4 o;
        o.x = (xv.x > 0.0f) ? e1.x : e0.x;
        o.y = (xv.y > 0.0f) ? e1.y : e0.y;
        o.z = (xv.z > 0.0f) ? e1.z : e0.z;
        o.w = (xv.w > 0.0f) ? e1.w : e0.w;
        __builtin_nontemporal_store(o, (f4*)(out + off));
    }
}

extern "C" void kernel_launch(void* const* d_in, const int* in_sizes, int n_in,
                              void* d_out, int out_size, void* d_ws, size_t ws_size,
                              hipStream_t stream)
{
    const float* x    = (const float*)d_in[0];  // (8,4096,1024) f32
    const float* emb0 = (const float*)d_in[1];  // (1,1,1024)   f32
    const float* emb1 = (const float*)d_in[2];  // (1,1,1024)   f32
    float* out        = (float*)d_out;          // (8,4096,1024) f32
    (void)in_sizes; (void)n_in; (void)out_size; (void)d_ws; (void)ws_size;

    rosa_select_kernel<<<ROSA_BLOCKS, ROSA_THREADS, 0, stream>>>(x, emb0, emb1, out);
}